// FFT_18133351923904
// MI455X (gfx1250) — compile-verified
//
#include <hip/hip_runtime.h>
#include <math.h>

// ---------------------------------------------------------------------------
// Batched 64-pt complex FFT + fftshift as one real GEMM on CDNA5 WMMA.
//
//   out[row] = in[row] @ B,   B is 128x128 f32, constant:
//     m = j>>1 (input sample), k = c>>1 (freq bin), theta = 2*pi*(m*k mod 64)/64
//     sg = (-1)^m   (fftshift: roll by 32 <=> modulate input by (-1)^m)
//     B[2m  ][2k  ] =  sg*cos   B[2m+1][2k  ] =  sg*sin
//     B[2m  ][2k+1] = -sg*sin   B[2m+1][2k+1] =  sg*cos
//
// Per workgroup (256 thr = 8 waves): 128 rows. Each wave owns a 16-column
// N-strip of B in 64 VGPRs (32 k-steps x v2f) and runs two interleaved
// 16-row accumulator chains of V_WMMA_F32_16X16X4_F32.
// ---------------------------------------------------------------------------

typedef __attribute__((ext_vector_type(2))) float v2f;
typedef __attribute__((ext_vector_type(8))) float v8f;

#define ROWS_TOTAL   (4096 * 64)   // 262144 FFT rows
#define ROW_FLOATS   128           // 64 complex, interleaved
#define ROWS_PER_WG  128           // 8 row-blocks of 16
#define WG_COUNT     (ROWS_TOTAL / ROWS_PER_WG)  // 2048

__global__ __launch_bounds__(256) void fft64_shift_wmma(
    const float* __restrict__ in, float* __restrict__ out) {
  // 64-entry twiddle table: tab[t] = {cos(2*pi*t/64), sin(2*pi*t/64)}
  __shared__ v2f tab[64];

  const int tid = threadIdx.x;
  if (tid < 64) {
    float th = (float)tid * (float)(6.283185307179586476925286766559 / 64.0);
    float s, c;
    __sincosf(th, &s, &c);
    v2f t; t.x = c; t.y = s;
    tab[tid] = t;
  }
  __syncthreads();

  const int lane = tid & 31;
  const int wave = tid >> 5;       // 0..7 -> which 16-column strip of B
  const int ln   = lane & 15;      // N within strip / A row within block
  const int lh   = lane >> 4;      // lane half: K offset (+2) per WMMA layout

  // This lane's output column (0..127 of the interleaved row)
  const int col  = (wave << 4) | ln;
  const int kf   = col >> 1;       // frequency bin
  const int cpar = col & 1;        // 0 = real column, 1 = imag column

  // -------- build the B N-strip in registers: 32 k-steps x 2 VGPRs --------
  // WMMA B (4x16) layout assumption (mirrors documented A 16x4 layout):
  //   VGPR0: lanes0-15 hold K=4s+0, lanes16-31 hold K=4s+2
  //   VGPR1: lanes0-15 hold K=4s+1, lanes16-31 hold K=4s+3
  // => this lane supplies rows j0 = 2m, j1 = 2m+1 with m = 2s + lh.
  v2f Bf[32];
#pragma unroll
  for (int s = 0; s < 32; ++s) {
    int m = 2 * s + lh;
    v2f cs = tab[(m * kf) & 63];
    float sg = (m & 1) ? -1.0f : 1.0f;
    float c = sg * cs.x;
    float sn = sg * cs.y;
    v2f b;
    b.x = cpar ? -sn : c;   // row 2m   (real input sample)
    b.y = cpar ?  c  : sn;  // row 2m+1 (imag input sample)
    Bf[s] = b;
  }

  // -------- GEMM over this workgroup's 128 rows --------
  const size_t wg_row0 = (size_t)blockIdx.x * ROWS_PER_WG;
  const v2f* inv = (const v2f*)in;  // row r starts at v2f index r*64

#pragma unroll 1
  for (int bp = 0; bp < 8; bp += 2) {  // two 16-row blocks per iteration
    const size_t r0 = wg_row0 + (size_t)bp * 16 + (size_t)ln;
    const size_t r1 = r0 + 16;
    // A 16x4 layout: lanes0-15 row=ln K={4s,4s+1}; lanes16-31 K={4s+2,4s+3}
    const v2f* A0 = inv + r0 * 64 + lh;
    const v2f* A1 = inv + r1 * 64 + lh;

    v8f acc0 = {};
    v8f acc1 = {};
#pragma unroll
    for (int s = 0; s < 32; ++s) {
      v2f a0 = A0[2 * s];
      v2f a1 = A1[2 * s];
      acc0 = __builtin_amdgcn_wmma_f32_16x16x4_f32(
          false, a0, false, Bf[s], (short)0, acc0, false, false);
      acc1 = __builtin_amdgcn_wmma_f32_16x16x4_f32(
          false, a1, false, Bf[s], (short)0, acc1, false, false);
    }

    // D 16x16 layout: VGPR i, lanes0-15 -> row i, lanes16-31 -> row i+8; N=ln.
    const size_t orow0 = wg_row0 + (size_t)bp * 16 + (size_t)(lh * 8);
    float* o0 = out + orow0 * ROW_FLOATS + col;
    float* o1 = out + (orow0 + 16) * ROW_FLOATS + col;
#pragma unroll
    for (int i = 0; i < 8; ++i) {
      o0[(size_t)i * ROW_FLOATS] = acc0[i];
      o1[(size_t)i * ROW_FLOATS] = acc1[i];
    }
  }
}

extern "C" void kernel_launch(void* const* d_in, const int* in_sizes, int n_in,
                              void* d_out, int out_size, void* d_ws, size_t ws_size,
                              hipStream_t stream) {
  (void)in_sizes; (void)n_in; (void)d_ws; (void)ws_size; (void)out_size;
  const float* in = (const float*)d_in[0];
  float* out = (float*)d_out;
  fft64_shift_wmma<<<WG_COUNT, 256, 0, stream>>>(in, out);
}